// Cowan_network_15023795601949
// MI455X (gfx1250) — compile-verified
//
#include <hip/hip_runtime.h>
#include <hip/hip_bf16.h>
#include <math.h>

// ---------------------------------------------------------------------------
// Wilson-Cowan network on MI455X (gfx1250).
//   SIZE=512, N_ATT=4, BATCH=4096, TMAX=200.
//   Dominant work: 200 x [4096,512]@[512,512] f32 matmuls = 4.3e11 FLOP.
//   HBM traffic ~9MB total (AT + x0 + out); AT resident in L2 -> compute bound.
//   => run matmuls on the f32 matrix pipe: V_WMMA_F32_16X16X4_F32.
//   Batch rows are independent -> each block owns 16 rows, keeps x/y in LDS
//   for all 200 steps, syncing only with block barriers.
//   Elementwise tanh uses the CDNA5 hardware V_TANH_F32 transcendental
//   (co-executes with XDL WMMA ops) instead of the divergent OCML expansion.
// ---------------------------------------------------------------------------

typedef __attribute__((ext_vector_type(2))) float v2f;
typedef __attribute__((ext_vector_type(8))) float v8f;

#define WC_SIZE   512
#define WC_TMAX   200
#define WC_XPAD   516     // padded LDS row stride (floats): kills bank conflicts
#define WC_DX     0.04419417382415922f  // 1/sqrt(512)

// Wilson-Cowan constants
#define C_WEE 7.2f
#define C_WEI 2.0f
#define C_WIE 0.0f
#define C_WII 1.0f
#define C_AE  1.5f
#define C_AI  0.4f
#define C_HE  (-1.2f)
#define C_HI  0.1f
#define C_FE1 0.25f
#define C_FE2 0.65f
#define C_FI1 0.5f
#define C_FI2 0.5f
#define C_B1  3.7f
#define C_B2  1.0f
#define C_DT  0.1f

// Hardware tanh: V_TANH_F32 is a CDNA5 TRANS32 op (co-executes with WMMA).
#if defined(__has_builtin)
#if __has_builtin(__builtin_amdgcn_tanhf)
#define WC_TANH(x) __builtin_amdgcn_tanhf(x)
#endif
#endif
#ifndef WC_TANH
static __device__ __forceinline__ float wc_tanh_asm(float x) {
    float r;
    asm("v_tanh_f32 %0, %1" : "=v"(r) : "v"(x));
    return r;
}
#define WC_TANH(x) wc_tanh_asm(x)
#endif

// ---------------------------------------------------------------------------
// Kernel 1: build augmented matrix [B | I] (512 x 1024) and clipped eig vector.
//   B[:,0:508] = base_train, B[:,508:512] = base_fix.
// ---------------------------------------------------------------------------
__global__ __launch_bounds__(256) void wc_build_aug(
    const float* __restrict__ bt, const float* __restrict__ bf,
    const float* __restrict__ atr, const float* __restrict__ afx,
    float* __restrict__ aug, float* __restrict__ eig)
{
    int idx = blockIdx.x * 256 + threadIdx.x;   // 512*1024 elements
    int i = idx >> 10;
    int j = idx & 1023;
    float v;
    if (j < 508)       v = bt[i * 508 + j];
    else if (j < 512)  v = bf[i * 4 + (j - 508)];
    else               v = ((j - 512) == i) ? 1.0f : 0.0f;
    aug[idx] = v;
    if (idx < WC_SIZE) {
        float e = (idx < 508) ? atr[idx] : afx[idx - 508];
        eig[idx] = fminf(20.0f, fmaxf(-1e6f, e));
    }
}

// ---------------------------------------------------------------------------
// Kernel 2: Gauss-Jordan inversion (no pivoting; base is near-orthogonal).
//   Single workgroup of 1024 threads; thread t owns column t of the 512x1024
//   augmented matrix. Per pivot: normalize row k, broadcast the pivot column
//   through LDS, then a fully coalesced rank-1 sweep. O(n^3)=2.7e8 FLOP.
//   On exit aug[:,512:1024] = B^-1.
// ---------------------------------------------------------------------------
__global__ __launch_bounds__(1024) void wc_gauss_jordan(float* __restrict__ aug)
{
    __shared__ float fcol[WC_SIZE];
    __shared__ float sInvPiv;
    const int tid = threadIdx.x;
    for (int k = 0; k < WC_SIZE; ++k) {
        if (tid == 0) sInvPiv = 1.0f / aug[(size_t)k * 1024 + k];
        __syncthreads();
        float invPiv = sInvPiv;
        // pivot column snapshot (fcol[k]=0 makes row k a no-op in the sweep)
        if (tid < WC_SIZE) fcol[tid] = (tid == k) ? 0.0f : aug[(size_t)tid * 1024 + k];
        // normalize pivot row; this thread's column value stays in a register
        float rk = aug[(size_t)k * 1024 + tid] * invPiv;
        aug[(size_t)k * 1024 + tid] = rk;
        __syncthreads();
        // eliminate: column-owner sweep, coalesced 4KB rows
        #pragma unroll 4
        for (int i = 0; i < WC_SIZE; ++i) {
            float* p = aug + (size_t)i * 1024 + tid;
            *p = fmaf(-fcol[i], rk, *p);
        }
        __syncthreads();
    }
}

// ---------------------------------------------------------------------------
// Kernel 3: AT[k][n] = sum_j Binv[j][k] * base[n][j] * eig[j]  (= (B D B^-1)^T)
// written directly in WMMA-B-fragment order:
//   ATp[((kk*32 + tile)*32 + lane)*2 + comp]
//   kk=k>>2, tile=n>>4, lane=(n&15)+16*((k>>1)&1), comp=k&1
// so the sim kernel's B-frag load is one contiguous 256B global_load_b64/wave.
// ---------------------------------------------------------------------------
__global__ __launch_bounds__(256) void wc_build_at(
    const float* __restrict__ bt, const float* __restrict__ bf,
    const float* __restrict__ eig, const float* __restrict__ aug,
    float* __restrict__ atp)
{
    int k = blockIdx.x * 16 + threadIdx.x;   // coalesced over Binv columns
    int n = blockIdx.y * 16 + threadIdx.y;
    float s = 0.0f;
    for (int j = 0; j < 508; ++j)
        s = fmaf(aug[(size_t)j * 1024 + 512 + k], bt[n * 508 + j] * eig[j], s);
    for (int j = 508; j < 512; ++j)
        s = fmaf(aug[(size_t)j * 1024 + 512 + k], bf[n * 4 + (j - 508)] * eig[j], s);
    int kk   = k >> 2;
    int tile = n >> 4;
    int lane = (n & 15) + (((k >> 1) & 1) << 4);
    int comp = k & 1;
    atp[(((size_t)kk * 32 + tile) * 32 + lane) * 2 + comp] = s;
}

// ---------------------------------------------------------------------------
// Kernel 4: full simulation + readout. One block = 16 batch rows, 8 waves.
//   Wave w owns columns [w*64, w*64+64) as 4 WMMA tiles (32 acc VGPRs).
//   LDS holds the x strip (padded) and y strip for all 200 steps.
// ---------------------------------------------------------------------------
__global__ __launch_bounds__(256) void wc_sim(
    const float* __restrict__ x0, const float* __restrict__ att,
    const float* __restrict__ gamma_p, const float* __restrict__ atp,
    float* __restrict__ out)
{
    __shared__ __align__(16) float xs[16 * WC_XPAD];
    __shared__ __align__(16) float ys[16 * WC_SIZE];

    const int tid  = threadIdx.x;
    const int r0   = blockIdx.x * 16;
    const int lane = tid & 31;
    const int w    = tid >> 5;          // wave id 0..7
    const int m    = lane & 15;         // A-frag row / D-frag column
    const int kh   = lane >> 4;         // A-frag K-half / D-frag row-half

    // load this block's 16 rows of x; y starts equal to x
    for (int idx = tid; idx < 16 * WC_SIZE; idx += 256) {
        int r = idx >> 9, c = idx & 511;
        float v = x0[(size_t)(r0 + r) * WC_SIZE + c];
        xs[r * WC_XPAD + c] = v;
        ys[r * WC_SIZE + c] = v;
    }
    const float dtg = C_DT / gamma_p[0];
    __syncthreads();

    const float* bwave = atp + ((size_t)w * 128 + (size_t)lane * 2); // + kk*1024 + t*32*2
    v8f acc[4];

    for (int step = 0; step < WC_TMAX; ++step) {
        v8f z = {};
        acc[0] = z; acc[1] = z; acc[2] = z; acc[3] = z;

        // S = x @ AT over K=512 in chunks of 4 (V_WMMA_F32_16X16X4_F32).
        // unroll 8 -> 32 outstanding global_load_b64 to cover L2 latency.
        #pragma unroll 8
        for (int kk = 0; kk < 128; ++kk) {
            int kb = kk * 4 + kh * 2;
            v2f a = *(const v2f*)(xs + m * WC_XPAD + kb);           // ds_load_b64
            const float* bp = bwave + (size_t)kk * 1024;
            v2f b0 = *(const v2f*)(bp +   0);                        // coalesced
            v2f b1 = *(const v2f*)(bp +  64);
            v2f b2 = *(const v2f*)(bp + 128);
            v2f b3 = *(const v2f*)(bp + 192);
            acc[0] = __builtin_amdgcn_wmma_f32_16x16x4_f32(false, a, false, b0, (short)0, acc[0], false, false);
            acc[1] = __builtin_amdgcn_wmma_f32_16x16x4_f32(false, a, false, b1, (short)0, acc[1], false, false);
            acc[2] = __builtin_amdgcn_wmma_f32_16x16x4_f32(false, a, false, b2, (short)0, acc[2], false, false);
            acc[3] = __builtin_amdgcn_wmma_f32_16x16x4_f32(false, a, false, b3, (short)0, acc[3], false, false);
        }
        __syncthreads();   // all matmul reads of xs done before updates

        // elementwise Wilson-Cowan update; D-tile lane layout -> (b,n) owner
        const int nbase = w * 64 + m;
        const int rhalf = kh * 8;
        #pragma unroll
        for (int t = 0; t < 4; ++t) {
            int n = nbase + t * 16;
            #pragma unroll
            for (int j = 0; j < 8; ++j) {
                int b = rhalf + j;
                float S  = acc[t][j];
                float xv = xs[b * WC_XPAD + n];
                float yv = ys[b * WC_SIZE + n];
                float I1 = C_WEE * xv - C_WEI * yv + C_HE + WC_DX * S;
                float I2 = C_WIE * xv - C_WII * yv + C_HI;
                float fe = C_FE1 * WC_TANH(C_B1 * I1) + C_FE2;
                float fi = C_FI1 * WC_TANH(C_B2 * I2) + C_FI2;
                float xn = xv + C_DT * (-C_AE * xv + (1.0f - xv) * fe);
                float yn = yv + dtg  * (-C_AI * yv + (1.0f - yv) * fi);
                xs[b * WC_XPAD + n] = fminf(1.0f, fmaxf(0.0f, xn));
                ys[b * WC_SIZE + n] = fminf(1.0f, fmaxf(0.0f, yn));
            }
        }
        __syncthreads();   // updates visible before next step's matmul
    }

    // ---- readout: wave w reduces rows 2w and 2w+1 ----
    #pragma unroll
    for (int rr = 0; rr < 2; ++rr) {
        int b = w * 2 + rr;
        float X2 = 0.f, d0 = 0.f, d1 = 0.f, d2 = 0.f, d3 = 0.f;
        float A0 = 0.f, A1 = 0.f, A2s = 0.f, A3 = 0.f;
        for (int i = lane; i < WC_SIZE; i += 32) {
            float xv = xs[b * WC_XPAD + i];
            X2 += xv * xv;
            float a0 = att[0 * WC_SIZE + i], a1 = att[1 * WC_SIZE + i];
            float a2 = att[2 * WC_SIZE + i], a3 = att[3 * WC_SIZE + i];
            float t0 = a0 - xv, t1 = a1 - xv, t2 = a2 - xv, t3 = a3 - xv;
            d0 += t0 * t0; d1 += t1 * t1; d2 += t2 * t2; d3 += t3 * t3;
            A0 += a0 * a0; A1 += a1 * a1; A2s += a2 * a2; A3 += a3 * a3;
        }
        #pragma unroll
        for (int off = 16; off > 0; off >>= 1) {
            X2 += __shfl_xor(X2, off, 32);
            d0 += __shfl_xor(d0, off, 32);  d1 += __shfl_xor(d1, off, 32);
            d2 += __shfl_xor(d2, off, 32);  d3 += __shfl_xor(d3, off, 32);
            A0 += __shfl_xor(A0, off, 32);  A1 += __shfl_xor(A1, off, 32);
            A2s += __shfl_xor(A2s, off, 32); A3 += __shfl_xor(A3, off, 32);
        }
        if (lane == 0) {
            float s0 = sqrtf(A0 * X2) / d0;
            float s1 = sqrtf(A1 * X2) / d1;
            float s2 = sqrtf(A2s * X2) / d2;
            float s3 = sqrtf(A3 * X2) / d3;
            out[r0 + b] = s0 / (s0 + s1 + s2 + s3);
        }
    }
}

// ---------------------------------------------------------------------------
extern "C" void kernel_launch(void* const* d_in, const int* in_sizes, int n_in,
                              void* d_out, int out_size, void* d_ws, size_t ws_size,
                              hipStream_t stream)
{
    const float* x   = (const float*)d_in[0];  // [4096,512]
    const float* bt  = (const float*)d_in[1];  // [512,508]
    const float* bf  = (const float*)d_in[2];  // [512,4]
    const float* atr = (const float*)d_in[3];  // [508]
    const float* afx = (const float*)d_in[4];  // [4]
    const float* att = (const float*)d_in[5];  // [4,512]
    const float* gam = (const float*)d_in[6];  // scalar
    float* out = (float*)d_out;

    char* ws = (char*)d_ws;
    float* aug = (float*)ws;                     // 512*1024*4 = 2 MB
    float* eig = (float*)(ws + 2097152);         // 2 KB
    float* atp = (float*)(ws + 2099200);         // 512*512*4 = 1 MB

    int batch = in_sizes[0] / WC_SIZE;           // 4096

    wc_build_aug<<<(512 * 1024) / 256, 256, 0, stream>>>(bt, bf, atr, afx, aug, eig);
    wc_gauss_jordan<<<1, 1024, 0, stream>>>(aug);
    wc_build_at<<<dim3(32, 32), dim3(16, 16), 0, stream>>>(bt, bf, eig, aug, atp);
    wc_sim<<<batch / 16, 256, 0, stream>>>(x, att, gam, atp, out);
}